// model_69982197121789
// MI455X (gfx1250) — compile-verified
//
#include <hip/hip_runtime.h>

// ---------------------------------------------------------------------------
// 16-layer fp32 MLP (8 -> 14x10 -> 1) on 2M rows, mapped onto
// V_WMMA_F32_16X16X4_F32.
//
// D = A x B + C with A = augmented weights (16x16, K chunks of 4),
// B = activations (batch tile of 16 rows in the lanes). The WMMA D layout
// (VGPR p: lanes 0-15 = row M=p, lanes 16-31 = row M=p+8) is reused directly
// as the next layer's B operand: B-VGPR pair (D[2c], D[2c+1]) supplies
// K-rows (2c, 2c+1, 2c+8, 2c+9), mirroring the documented A-matrix layout
// (VGPR0 = K0 lo / K2 hi, VGPR1 = K1 lo / K3 hi). Bias is folded in as
// augmented feature 15 (held == 1.0 through all layers).
// ReLU = v_med3_f32(x, 0, +inf): one VALU op per element on the D->B path.
// ---------------------------------------------------------------------------

typedef __attribute__((ext_vector_type(2))) float v2f;
typedef __attribute__((ext_vector_type(8))) float v8f;

struct WPtrs {
    const float* W[16];
    const float* b[16];
};

// Pack weights into the per-lane A-matrix image:
//   packed[(i*32 + lane)*4 + t] = A_vgpr[4*i + t][lane],  i=0..31, t=0..3
// VGPR index j = 4i+t:  layer l = j/8, chunk c = (j%8)/2, which = j%2.
//   A_vgpr(2c+which)[lane] = W_aug[lane&15, 2c + which + (lane<16 ? 0 : 8)]
// W_aug[m,k]: real weight for m<out_dim,k<in_dim; b[m] at k==15 (m<out_dim);
//             1.0 at (15,15) to carry the bias feature; 0 elsewhere.
__global__ __launch_bounds__(256) void prep_weights(WPtrs P, float* __restrict__ packed) {
    int t = blockIdx.x * blockDim.x + threadIdx.x;   // 0..4095
    if (t >= 16 * 8 * 32) return;
    int t4   = t & 3;           // float within the b128 load
    int lane = (t >> 2) & 31;   // lane
    int i    = t >> 7;          // load-instruction index 0..31
    int j    = i * 4 + t4;      // flat weight-VGPR index 0..127
    int l     = j >> 3;         // layer 0..15
    int v8    = j & 7;
    int chunk = v8 >> 1;
    int which = v8 & 1;
    int m = lane & 15;
    int k = 2 * chunk + which + ((lane < 16) ? 0 : 8);
    int in_dim  = (l == 0) ? 8 : 10;
    int out_dim = (l == 15) ? 1 : 10;
    float val = 0.0f;
    if (k == 15) {
        if (m < out_dim)      val = P.b[l][m];
        else if (m == 15)     val = 1.0f;      // bias-carry identity
    } else if (m < out_dim && k < in_dim) {
        val = P.W[l][m * in_dim + k];
    }
    packed[t] = val;
}

__global__ __launch_bounds__(256) void mlp_wmma_kernel(const float* __restrict__ x,
                                                       const float* __restrict__ packed,
                                                       float* __restrict__ out,
                                                       int nrows) {
    const int lane = threadIdx.x & 31;
    const int wave = blockIdx.x * (blockDim.x >> 5) + (threadIdx.x >> 5);
    const long base = (long)wave * 16;
    if (base >= nrows) return;                       // whole-wave exit: EXEC stays all-ones

    // ---- load this wave's 16x8 input tile (clamped; hi lanes discarded) ----
    long row  = base + (lane & 15);
    long rld  = (row < nrows) ? row : (long)(nrows - 1);
    const float4* xp = (const float4*)(x + rld * 8);
    float4 x0 = xp[0];
    float4 x1 = xp[1];
    const bool lo = lane < 16;

    // Augmented input: features 0-7 = x, 8-14 = 0, 15 = 1 (bias carrier).
    v8f acts;
    acts[0] = lo ? x0.x : 0.0f;
    acts[1] = lo ? x0.y : 0.0f;
    acts[2] = lo ? x0.z : 0.0f;
    acts[3] = lo ? x0.w : 0.0f;
    acts[4] = lo ? x1.x : 0.0f;
    acts[5] = lo ? x1.y : 0.0f;
    acts[6] = lo ? x1.z : 0.0f;
    acts[7] = lo ? x1.w : 1.0f;

    // ---- pull the 16KB permuted weight image into 128 VGPRs ----
    float4 w[32];
    const float4* wp = (const float4*)packed + lane;
#pragma unroll
    for (int i = 0; i < 32; ++i) w[i] = wp[i * 32];

    const float inf = __builtin_inff();

    // ---- 16 layers x 4 accumulating WMMAs ----
#pragma unroll
    for (int l = 0; l < 16; ++l) {
        v8f c = {0.f, 0.f, 0.f, 0.f, 0.f, 0.f, 0.f, 0.f};
#pragma unroll
        for (int ch = 0; ch < 4; ++ch) {
            const int j = l * 8 + ch * 2;            // even flat index
            const float4 q = w[j >> 2];
            v2f a;
            if (j & 2) { a[0] = q.z; a[1] = q.w; }
            else       { a[0] = q.x; a[1] = q.y; }
            v2f bb;
            bb[0] = acts[2 * ch];
            bb[1] = acts[2 * ch + 1];
            c = __builtin_amdgcn_wmma_f32_16x16x4_f32(
                    /*neg_a=*/false, a, /*neg_b=*/false, bb,
                    /*c_mod=*/(short)0, c, /*reuse_a=*/false, /*reuse_b=*/false);
        }
        if (l == 0) {
            acts = c;                                // layer 0: no ReLU
        } else {
            // ReLU as a single v_med3_f32: clamp to [0, +inf)
#pragma unroll
            for (int p = 0; p < 8; ++p)
                acts[p] = __builtin_amdgcn_fmed3f(c[p], 0.0f, inf);
        }
    }

    // Final scalar output = D row 0 (already ReLU'd), valid in lanes 0-15.
    if (lo && row < nrows) out[row] = acts[0];
}

extern "C" void kernel_launch(void* const* d_in, const int* in_sizes, int n_in,
                              void* d_out, int out_size, void* d_ws, size_t ws_size,
                              hipStream_t stream) {
    const float* x = (const float*)d_in[0];
    WPtrs P;
    for (int i = 0; i < 16; ++i) {
        P.W[i] = (const float*)d_in[1 + i];
        P.b[i] = (const float*)d_in[17 + i];
    }
    float* packed = (float*)d_ws;                    // 4096 floats = 16 KB
    float* out    = (float*)d_out;
    const int nrows = in_sizes[0] / 8;

    prep_weights<<<16, 256, 0, stream>>>(P, packed);

    const int tiles  = (nrows + 15) / 16;            // 16 rows per wave
    const int blocks = (tiles + 7) / 8;              // 8 waves per 256-thread block
    mlp_wmma_kernel<<<blocks, 256, 0, stream>>>(x, packed, out, nrows);
}